// RetentionMedianValues_48756468744806
// MI455X (gfx1250) — compile-verified
//
#include <hip/hip_runtime.h>

#define KSZ    5
#define PAD    2
#define W_IMG  256
#define H_IMG  256
#define TILE_W 64
#define TILE_H 16
#define HALO_X 4                         // left halo enlarged to 4 floats for 16B alignment
#define LDS_W  (TILE_W + 2 * HALO_X)     // 72 floats per row (288B, 16B-aligned stride)
#define LDS_H  (TILE_H + 2 * PAD)        // 20
#define NCHUNK128 ((LDS_W / 4) * LDS_H)  // 18*20 = 360 16-byte chunks
#define NSLOT     (LDS_W * LDS_H)        // 1440 floats

// compare-exchange: a=min, b=max  -> v_min_num_f32 / v_max_num_f32
__device__ __forceinline__ void cas(float& a, float& b) {
  float t = fminf(a, b);
  b = fmaxf(a, b);
  a = t;
}

// Partial network over v[0..N-1]: global min -> v[0], global max -> v[N-1].
// Middle elements survive in v[1..N-2] (order unspecified).
template <int N>
__device__ __forceinline__ void mnmx(float* v) {
  constexpr int M = (N & 1) ? (N - 1) : N;
  if constexpr (N & 1) cas(v[0], v[N - 1]);
#pragma unroll
  for (int i = 0; i < M; i += 2) cas(v[i], v[i + 1]);      // mins even, maxes odd
#pragma unroll
  for (int i = 2; i < M; i += 2) cas(v[0], v[i]);          // min chain -> v[0]
#pragma unroll
  for (int i = 1; i + 2 < M; i += 2) cas(v[i], v[i + 2]);  // max chain -> v[M-1]
  if constexpr (N & 1) cas(v[M - 1], v[N - 1]);
}

__global__ __launch_bounds__(256) void median5x5_kernel(const float* __restrict__ in,
                                                        float* __restrict__ out) {
  __shared__ float tile[LDS_H][LDS_W];

  const int tid = threadIdx.x;
  const int x0  = blockIdx.x * TILE_W;
  const int y0  = blockIdx.y * TILE_H;
  const int z   = blockIdx.z;  // flattened (n, c) plane
  const float* plane = in + (size_t)z * (H_IMG * W_IMG);

  // Uniform per-block predicate: whole (enlarged) halo in-bounds -> pure async path.
  const bool interior = (x0 >= HALO_X) && (x0 + TILE_W + HALO_X <= W_IMG) &&
                        (y0 >= PAD) && (y0 + TILE_H + PAD <= H_IMG);

  if (interior) {
    // Fill 72x20 LDS tile with async global->LDS DMA, 16 bytes per lane-op.
    // x0 is a multiple of 64, so (x0 - HALO_X) is 16B-aligned; LDS row stride
    // is 288B and lx is a multiple of 4 floats -> both sides 16B-aligned.
#pragma unroll
    for (int it = 0; it < 2; ++it) {
      const int i = tid + it * 256;
      if (i < NCHUNK128) {
        const int ly = i / (LDS_W / 4);
        const int lx = (i % (LDS_W / 4)) * 4;
        // Low 32 bits of a generic LDS address are the LDS byte offset.
        unsigned ldsOff = (unsigned)(unsigned long long)&tile[ly][lx];
        unsigned long long gaddr =
            (unsigned long long)(plane + (size_t)(y0 - PAD + ly) * W_IMG + (x0 - HALO_X + lx));
        asm volatile("global_load_async_to_lds_b128 %0, %1, off"
                     :: "v"(ldsOff), "v"(gaddr)
                     : "memory");
      }
    }
#if __has_builtin(__builtin_amdgcn_s_wait_asynccnt)
    __builtin_amdgcn_s_wait_asynccnt(0);
#else
    asm volatile("s_wait_asynccnt 0" ::: "memory");
#endif
  } else {
    // Boundary tiles: scalar loads with zero padding.
    for (int i = tid; i < NSLOT; i += 256) {
      const int ly = i / LDS_W;
      const int lx = i % LDS_W;
      const int gy = y0 - PAD + ly;
      const int gx = x0 - HALO_X + lx;
      float val = 0.0f;
      if ((unsigned)gy < (unsigned)H_IMG && (unsigned)gx < (unsigned)W_IMG)
        val = plane[(size_t)gy * W_IMG + gx];
      tile[ly][lx] = val;
    }
  }
  __syncthreads();

  const int tx  = tid & (TILE_W - 1);
  const int tyq = tid >> 6;  // 0..3, each thread does 4 consecutive rows
  float* outPlane = out + (size_t)z * (H_IMG * W_IMG);

#pragma unroll
  for (int k = 0; k < 4; ++k) {
    const int ty = tyq * 4 + k;
    float v[14];
    // Window for output (y0+ty, x0+tx): rows ty..ty+4, cols tx+2..tx+6
    // (tile x origin is x0 - HALO_X = x0 - 4, window starts at x0+tx-2).
#define E(w) tile[ty + (w) / 5][tx + (HALO_X - PAD) + (w) % 5]
#pragma unroll
    for (int w = 0; w < 14; ++w) v[w] = E(w);
    // Forgetful exact median-of-25 (lower median == true median, rank 12):
    // min of any 14-subset has rank <= 11, max has rank >= 13 -> discard both.
    mnmx<14>(v);
    v[0] = E(14); mnmx<13>(v);
    v[0] = E(15); mnmx<12>(v);
    v[0] = E(16); mnmx<11>(v);
    v[0] = E(17); mnmx<10>(v);
    v[0] = E(18); mnmx<9>(v);
    v[0] = E(19); mnmx<8>(v);
    v[0] = E(20); mnmx<7>(v);
    v[0] = E(21); mnmx<6>(v);
    v[0] = E(22); mnmx<5>(v);
    v[0] = E(23); mnmx<4>(v);
    v[0] = E(24);
#undef E
    cas(v[0], v[1]);
    cas(v[1], v[2]);
    cas(v[0], v[1]);  // v[1] = median of final 3 survivors = median of 25
    outPlane[(size_t)(y0 + ty) * W_IMG + (x0 + tx)] = v[1];
  }
}

extern "C" void kernel_launch(void* const* d_in, const int* in_sizes, int n_in,
                              void* d_out, int out_size, void* d_ws, size_t ws_size,
                              hipStream_t stream) {
  (void)n_in; (void)out_size; (void)d_ws; (void)ws_size;
  const float* x = (const float*)d_in[0];
  float* out = (float*)d_out;
  const int planes = in_sizes[0] / (H_IMG * W_IMG);  // N*C = 256
  dim3 grid(W_IMG / TILE_W, H_IMG / TILE_H, planes);
  dim3 block(256);
  hipLaunchKernelGGL(median5x5_kernel, grid, block, 0, stream, x, out);
}